// InferTree_40157944218107
// MI455X (gfx1250) — compile-verified
//
#include <hip/hip_runtime.h>

// Problem constants (from reference): N=8192, C=512, B1=B2=16, out cols = 257.
#define NROWS 8192
#define CDIM  512
#define OUTC  257

typedef __attribute__((ext_vector_type(2))) float v2f;
typedef __attribute__((ext_vector_type(8))) float v8f;

// ---- WMMA f32 16x16x4 fragment loads (wave32) ----
// A (16x4, MxK): lane L<16 -> row M=L, K={0,1}; lanes 16..31 -> K={2,3}.
__device__ __forceinline__ v2f ldA(const float* __restrict__ A, int lda,
                                   int row0, int k0, int lane) {
  int m  = lane & 15;
  int kk = (lane >> 4) << 1;
  return *reinterpret_cast<const v2f*>(A + (size_t)(row0 + m) * lda + (k0 + kk));
}
// B (4x16, KxN): VGPR0 = rows K=0 (lanes 0-15) / K=2 (lanes 16-31); VGPR1 = K=1 / K=3.
__device__ __forceinline__ v2f ldB(const float* __restrict__ B, int ldb,
                                   int k0, int n0, int lane) {
  int n  = lane & 15;
  int kk = (lane >> 4) << 1;
  v2f r;
  r.x = B[(size_t)(k0 + kk)     * ldb + (n0 + n)];
  r.y = B[(size_t)(k0 + kk + 1) * ldb + (n0 + n)];
  return r;
}

#define WMMA_F32(a, b, c) \
  __builtin_amdgcn_wmma_f32_16x16x4_f32(false, (a), false, (b), (short)0, (c), false, false)

// ---------------------------------------------------------------------------
// Kernel 0: pack fc2_w (16 x 17 x 512) rows c=0..15 into dense A0 (256 x 512),
// zero the 257th output column and the penalty scalar.
__global__ __launch_bounds__(256) void prep_kernel(const float* __restrict__ fc2_w,
                                                   float* __restrict__ A0,
                                                   float* __restrict__ out) {
  int tid = blockIdx.x * 256 + threadIdx.x;          // 512 blocks -> 131072 threads
  int j = tid >> 9;                                  // packed row 0..255
  int e = tid & 511;
  int b = j >> 4, c = j & 15;
  A0[tid] = fc2_w[((size_t)b * 17 + c) * CDIM + e];
  if (tid < NROWS) out[(size_t)tid * OUTC + 256] = 0.0f;
  if (tid == 0)    out[(size_t)NROWS * OUTC]     = 0.0f;   // penalty scalar
}

// ---------------------------------------------------------------------------
// Kernel 1: W2[b*16+c, d] = sum_e A0[b*16+c, e] * conv2_w[b, e, d]
// One block per b; 8 waves, each wave owns 64 of 512 output columns (4 tiles).
__global__ __launch_bounds__(256) void gemm_w2(const float* __restrict__ A0,
                                               const float* __restrict__ conv2_w,
                                               float* __restrict__ W2) {
  int b    = blockIdx.x;
  int lane = threadIdx.x & 31;
  int wave = threadIdx.x >> 5;
  int n0   = wave * 64;
  const float* A = A0      + (size_t)b * 16  * CDIM;   // 16 x 512 block
  const float* B = conv2_w + (size_t)b * CDIM * CDIM;  // 512 x 512 (e-major)

  v8f c0 = {}, c1 = {}, c2 = {}, c3 = {};
  #pragma unroll 2
  for (int k0 = 0; k0 < CDIM; k0 += 4) {
    v2f a  = ldA(A, CDIM, 0, k0, lane);
    v2f b0 = ldB(B, CDIM, k0, n0,      lane);
    v2f b1 = ldB(B, CDIM, k0, n0 + 16, lane);
    v2f b2 = ldB(B, CDIM, k0, n0 + 32, lane);
    v2f b3 = ldB(B, CDIM, k0, n0 + 48, lane);
    c0 = WMMA_F32(a, b0, c0);
    c1 = WMMA_F32(a, b1, c1);
    c2 = WMMA_F32(a, b2, c2);
    c3 = WMMA_F32(a, b3, c3);
  }
  int n  = lane & 15;
  int mo = (lane >> 4) << 3;
  #pragma unroll
  for (int v = 0; v < 8; ++v) {
    size_t r = (size_t)(b * 16 + v + mo) * CDIM;
    W2[r + n0      + n] = c0[v];
    W2[r + n0 + 16 + n] = c1[v];
    W2[r + n0 + 32 + n] = c2[v];
    W2[r + n0 + 48 + n] = c3[v];
  }
}

// ---------------------------------------------------------------------------
// Kernel 2: V = W2 (256x512) @ conv1_w (512x512), stored transposed:
// Vt[k, j] = sum_d W2[j, d] * conv1_w[d, k].   16 blocks x 8 waves = 128 waves;
// wave -> (row tile of 16, 64-col group) over the 256 x 512 result.
__global__ __launch_bounds__(256) void gemm_vt(const float* __restrict__ W2,
                                               const float* __restrict__ conv1_w,
                                               float* __restrict__ Vt) {
  int lane = threadIdx.x & 31;
  int gw   = blockIdx.x * 8 + (threadIdx.x >> 5);   // 0..127
  int row0 = (gw >> 3) * 16;                        // 0..240
  int n0   = (gw & 7) * 64;                         // 0..448

  v8f c0 = {}, c1 = {}, c2 = {}, c3 = {};
  #pragma unroll 2
  for (int k0 = 0; k0 < CDIM; k0 += 4) {
    v2f a  = ldA(W2, CDIM, row0, k0, lane);
    v2f b0 = ldB(conv1_w, CDIM, k0, n0,      lane);
    v2f b1 = ldB(conv1_w, CDIM, k0, n0 + 16, lane);
    v2f b2 = ldB(conv1_w, CDIM, k0, n0 + 32, lane);
    v2f b3 = ldB(conv1_w, CDIM, k0, n0 + 48, lane);
    c0 = WMMA_F32(a, b0, c0);
    c1 = WMMA_F32(a, b1, c1);
    c2 = WMMA_F32(a, b2, c2);
    c3 = WMMA_F32(a, b3, c3);
  }
  int n  = lane & 15;
  int mo = (lane >> 4) << 3;
  #pragma unroll
  for (int v = 0; v < 8; ++v) {
    int m = row0 + v + mo;
    Vt[(size_t)(n0      + n) * 256 + m] = c0[v];
    Vt[(size_t)(n0 + 16 + n) * 256 + m] = c1[v];
    Vt[(size_t)(n0 + 32 + n) * 256 + m] = c2[v];
    Vt[(size_t)(n0 + 48 + n) * 256 + m] = c3[v];
  }
}

// ---------------------------------------------------------------------------
// Kernel 3: out[n, j] = sum_k x[n, k] * Vt[k, j] + fc2_b[j/16, j%16]
// out row stride 257 (col 256 already zeroed). Block = 32 rows x 256 cols;
// wave = 16 rows x 64 cols (4 accumulator tiles -> 4 independent WMMA chains).
__global__ __launch_bounds__(256) void gemm_out(const float* __restrict__ x,
                                                const float* __restrict__ Vt,
                                                const float* __restrict__ fc2_b,
                                                float* __restrict__ out) {
  int lane = threadIdx.x & 31;
  int wave = threadIdx.x >> 5;
  int row0 = blockIdx.x * 32 + (wave >> 2) * 16;
  int n0   = (wave & 3) * 64;

  v8f c0 = {}, c1 = {}, c2 = {}, c3 = {};
  #pragma unroll 2
  for (int k0 = 0; k0 < CDIM; k0 += 4) {
    v2f a  = ldA(x, CDIM, row0, k0, lane);
    v2f b0 = ldB(Vt, 256, k0, n0,      lane);
    v2f b1 = ldB(Vt, 256, k0, n0 + 16, lane);
    v2f b2 = ldB(Vt, 256, k0, n0 + 32, lane);
    v2f b3 = ldB(Vt, 256, k0, n0 + 48, lane);
    c0 = WMMA_F32(a, b0, c0);
    c1 = WMMA_F32(a, b1, c1);
    c2 = WMMA_F32(a, b2, c2);
    c3 = WMMA_F32(a, b3, c3);
  }

  int n  = lane & 15;
  int mo = (lane >> 4) << 3;
  int j0 = n0 + n, j1 = n0 + 16 + n, j2 = n0 + 32 + n, j3 = n0 + 48 + n;
  float bias0 = fc2_b[(j0 >> 4) * 17 + (j0 & 15)];
  float bias1 = fc2_b[(j1 >> 4) * 17 + (j1 & 15)];
  float bias2 = fc2_b[(j2 >> 4) * 17 + (j2 & 15)];
  float bias3 = fc2_b[(j3 >> 4) * 17 + (j3 & 15)];
  #pragma unroll
  for (int v = 0; v < 8; ++v) {
    size_t r = (size_t)(row0 + v + mo) * OUTC;
    out[r + j0] = c0[v] + bias0;
    out[r + j1] = c1[v] + bias1;
    out[r + j2] = c2[v] + bias2;
    out[r + j3] = c3[v] + bias3;
  }
}

// ---------------------------------------------------------------------------
extern "C" void kernel_launch(void* const* d_in, const int* in_sizes, int n_in,
                              void* d_out, int out_size, void* d_ws, size_t ws_size,
                              hipStream_t stream) {
  // setup_inputs order: 0:x 1:labels 2:conv1_w 3:fc1_w 4:fc1_b 5:conv2_w 6:fc2_w 7:fc2_b
  const float* x       = (const float*)d_in[0];
  const float* conv1_w = (const float*)d_in[2];
  const float* conv2_w = (const float*)d_in[5];
  const float* fc2_w   = (const float*)d_in[6];
  const float* fc2_b   = (const float*)d_in[7];
  float* out = (float*)d_out;

  // Workspace: A0 (256x512) | W2 (256x512) | Vt (512x256)  = 1.5 MB total
  float* A0 = (float*)d_ws;
  float* W2 = A0 + 256 * CDIM;
  float* Vt = W2 + 256 * CDIM;

  prep_kernel<<<512, 256, 0, stream>>>(fc2_w, A0, out);
  gemm_w2 <<<16,  256, 0, stream>>>(A0, conv2_w, W2);
  gemm_vt <<<16,  256, 0, stream>>>(W2, conv1_w, Vt);
  gemm_out<<<256, 256, 0, stream>>>(x, Vt, fc2_b, out);
}